// QuantumSuperpositionPathLayer_90280212562552
// MI455X (gfx1250) — compile-verified
//
#include <hip/hip_runtime.h>
#include <math.h>

typedef __attribute__((ext_vector_type(2))) float v2f;
typedef __attribute__((ext_vector_type(8))) float v8f;

#define NQ  6
#define DIM 64

// ============================================================================
// Phase 1: build the fused circuit unitary W (64 x 128: [Re U^T | Im U^T])
// and the Z-expectation->output matrix M (64 x 4), once per call.
// One block, 64 threads; thread j evolves basis state e_j through the fixed
// 3x(Rot layer + CNOT ladder) circuit. State kept in LDS (pad 65 -> no bank
// conflicts: bank = (j + k) mod 64 for a fixed k).
// ============================================================================
__global__ __launch_bounds__(64) void build_unitary(
    const float* __restrict__ path_params,   // (3,6,3)
    const float* __restrict__ W_out,         // (4,6)
    float* __restrict__ ws)                  // [0..8191]=W, [8192..8447]=M
{
  __shared__ float sre[DIM * 65];
  __shared__ float sim[DIM * 65];
  const int j = threadIdx.x;                 // column index 0..63
  float* re = &sre[j * 65];
  float* im = &sim[j * 65];
  for (int k = 0; k < DIM; ++k) { re[k] = (k == j) ? 1.f : 0.f; im[k] = 0.f; }

  for (int layer = 0; layer < 3; ++layer) {
    for (int w = 0; w < NQ; ++w) {
      const float phi   = path_params[(layer*NQ + w)*3 + 0];
      const float theta = path_params[(layer*NQ + w)*3 + 1];
      const float omega = path_params[(layer*NQ + w)*3 + 2];
      float ct, st, ca, sa, cb, sb;
      __sincosf(0.5f * theta,         &st, &ct);
      __sincosf(0.5f * (phi + omega), &sa, &ca);
      __sincosf(0.5f * (phi - omega), &sb, &cb);
      // Rot = [[ep*ct, -em*st],[conj(em)*st, conj(ep)*ct]],
      // ep = ca - i*sa, em = cb + i*sb
      const float g00r =  ca*ct, g00i = -sa*ct;
      const float g01r = -cb*st, g01i = -sb*st;
      const float g10r =  cb*st, g10i = -sb*st;
      const float g11r =  ca*ct, g11i =  sa*ct;
      const int bw = 5 - w;                  // wire 0 = MSB
      for (int p = 0; p < 32; ++p) {
        const int k0 = ((p >> bw) << (bw + 1)) | (p & ((1 << bw) - 1));
        const int k1 = k0 | (1 << bw);
        const float a0r = re[k0], a0i = im[k0];
        const float a1r = re[k1], a1i = im[k1];
        re[k0] = g00r*a0r - g00i*a0i + g01r*a1r - g01i*a1i;
        im[k0] = g00r*a0i + g00i*a0r + g01r*a1i + g01i*a1r;
        re[k1] = g10r*a0r - g10i*a0i + g11r*a1r - g11i*a1i;
        im[k1] = g10r*a0i + g10i*a0r + g11r*a1i + g11i*a1r;
      }
    }
    for (int w = 0; w < NQ - 1; ++w) {       // CNOT(control=w, target=w+1)
      const int bt = 4 - w;                  // target bit; control bit = bt+1
      for (int p = 0; p < 16; ++p) {
        const int k0 = ((p >> bt) << (bt + 2)) | (2 << bt) | (p & ((1 << bt) - 1));
        const int k1 = k0 | (1 << bt);
        const float tr = re[k0], ti = im[k0];
        re[k0] = re[k1]; im[k0] = im[k1];
        re[k1] = tr;     im[k1] = ti;
      }
    }
  }
  // W row j = [Re U[0..63][j] | Im U[0..63][j]]  (B operand of the GEMM)
  for (int n = 0; n < DIM; ++n) {
    ws[j*128 + n]      = re[n];
    ws[j*128 + 64 + n] = im[n];
  }
  // M[k][p] = sum_w W_out[p][w] * (bit_{5-w}(k) ? -1 : +1)
  for (int p = 0; p < 4; ++p) {
    float acc = 0.f;
    for (int w = 0; w < NQ; ++w)
      acc += W_out[p*NQ + w] * (((j >> (5 - w)) & 1) ? -1.f : 1.f);
    ws[DIM*128 + j*4 + p] = acc;
  }
}

// ============================================================================
// Phase 2: per-row state build + (B x 64) @ (64 x 128) f32 WMMA GEMM +
// probability/output contraction. 256 threads = 8 waves; wave = 16 rows.
// ============================================================================
__global__ __launch_bounds__(256) void qsp_main(
    const float* __restrict__ x, const float* __restrict__ W_in,
    const float* __restrict__ b_in, const float* __restrict__ ws,
    const float* __restrict__ b_out, float* __restrict__ out, int Btot)
{
  __shared__ float ldsW[DIM * 128];          // 32 KB: staged U^T (Re|Im)
  const int tid = threadIdx.x;
  {
    const float4* src = (const float4*)ws;
    float4* dst = (float4*)ldsW;
    #pragma unroll
    for (int i = 0; i < 8; ++i) dst[tid + 256*i] = src[tid + 256*i];
  }
  __syncthreads();

  const int wave = tid >> 5;
  const int lane = tid & 31;
  const int lo = lane & 15, hi = lane >> 4;
  const long rbase = (long)blockIdx.x * 128 + wave * 16;
  const long row = rbase + lo;

  // Per-row RY product factors: G[q][b] for wire q, bit value b
  float G[NQ][2];
  {
    float xr[4] = {0.f, 0.f, 0.f, 0.f};
    if (row < Btot) {
      xr[0] = x[row*4+0]; xr[1] = x[row*4+1];
      xr[2] = x[row*4+2]; xr[3] = x[row*4+3];
    }
    #pragma unroll
    for (int q = 0; q < NQ; ++q) {
      float acc = b_in[q];
      #pragma unroll
      for (int p = 0; p < 4; ++p) acc += W_in[q*4+p] * xr[p];
      const float half = 0.78539816339744831f * tanhf(acc);  // (pi/4)*tanh
      float s, c;
      __sincosf(half, &s, &c);
      G[q][0] = c - s;
      G[q][1] = c + s;
    }
  }

  // A operand (16x4 f32 WMMA layout): lane holds S[row][4*kk + 2*hi + j]
  // S[k] = (1/8) * G0[k5] G1[k4] G2[k3] G3[k2] G4[k1] G5[k0]
  v2f areg[16];
  {
    const float t0 = 0.125f * G[4][hi] * G[5][0];
    const float t1 = 0.125f * G[4][hi] * G[5][1];
    float P3[8];
    #pragma unroll
    for (int i = 0; i < 8; ++i)
      P3[i] = G[0][(i >> 2) & 1] * G[1][(i >> 1) & 1] * G[2][i & 1];
    #pragma unroll
    for (int kk = 0; kk < 16; ++kk) {
      const float base = P3[kk >> 1] * G[3][kk & 1];
      areg[kk][0] = base * t0;
      areg[kk][1] = base * t1;
    }
  }

  float outacc[8][4];
  #pragma unroll
  for (int v = 0; v < 8; ++v)
    #pragma unroll
    for (int p = 0; p < 4; ++p) outacc[v][p] = 0.f;

  const float* Mrow = &ws[DIM * 128];

  for (int nt = 0; nt < 4; ++nt) {           // N-tiles: Re cols nt, Im cols nt+4
    v8f cr = {};
    v8f ci = {};
    const int colR = nt*16 + lo;
    const int colI = colR + 64;
    #pragma unroll
    for (int kk = 0; kk < 16; ++kk) {
      v2f br, bi;                            // B operand: rows striped by lane half
      br[0] = ldsW[(4*kk + hi    )*128 + colR];
      br[1] = ldsW[(4*kk + hi + 2)*128 + colR];
      bi[0] = ldsW[(4*kk + hi    )*128 + colI];
      bi[1] = ldsW[(4*kk + hi + 2)*128 + colI];
      cr = __builtin_amdgcn_wmma_f32_16x16x4_f32(false, areg[kk], false, br,
                                                 (short)0, cr, false, false);
      ci = __builtin_amdgcn_wmma_f32_16x16x4_f32(false, areg[kk], false, bi,
                                                 (short)0, ci, false, false);
    }
    // probs -> output contraction with M (C layout: VGPR v -> row v + 8*hi)
    const int n_amp = nt*16 + lo;
    const float m0 = Mrow[n_amp*4+0], m1 = Mrow[n_amp*4+1];
    const float m2 = Mrow[n_amp*4+2], m3 = Mrow[n_amp*4+3];
    #pragma unroll
    for (int v = 0; v < 8; ++v) {
      const float pr = cr[v]*cr[v] + ci[v]*ci[v];
      outacc[v][0] += pr * m0;
      outacc[v][1] += pr * m1;
      outacc[v][2] += pr * m2;
      outacc[v][3] += pr * m3;
    }
  }

  // Reduce over the 16-lane half-wave (the k-column dimension)
  #pragma unroll
  for (int v = 0; v < 8; ++v)
    #pragma unroll
    for (int p = 0; p < 4; ++p) {
      float s = outacc[v][p];
      s += __shfl_xor(s, 1);
      s += __shfl_xor(s, 2);
      s += __shfl_xor(s, 4);
      s += __shfl_xor(s, 8);
      outacc[v][p] = s;
    }

  if (lo < 4) {
    const int p = lo;
    const float bo = b_out[p];
    #pragma unroll
    for (int v = 0; v < 8; ++v) {
      const long r = rbase + v + 8*hi;
      if (r < Btot) out[r*4 + p] = outacc[v][p] + bo;
    }
  }
}

extern "C" void kernel_launch(void* const* d_in, const int* in_sizes, int n_in,
                              void* d_out, int out_size, void* d_ws, size_t ws_size,
                              hipStream_t stream) {
  (void)n_in; (void)out_size; (void)ws_size;
  const float* x     = (const float*)d_in[0];
  const float* W_in  = (const float*)d_in[1];
  const float* b_in  = (const float*)d_in[2];
  const float* pp    = (const float*)d_in[3];
  const float* W_out = (const float*)d_in[4];
  const float* b_out = (const float*)d_in[5];
  float* out = (float*)d_out;
  float* ws  = (float*)d_ws;                 // needs 8448 floats (33 KB)
  const int Btot = in_sizes[0] / 4;

  build_unitary<<<1, 64, 0, stream>>>(pp, W_out, ws);
  const int blocks = (Btot + 127) / 128;
  qsp_main<<<blocks, 256, 0, stream>>>(x, W_in, b_in, ws, b_out, out, Btot);
}